// Net_18734647345153
// MI455X (gfx1250) — compile-verified
//
#include <hip/hip_runtime.h>

// out = copy(A); out[index[i], :] += B[i, :]  (duplicate-accumulating scatter-add)
// N=100000 target rows, M=500000 source rows, D=128 fp32.
//
// Strategy:
//  - A -> out via async D2D memcpy (peak BW, capture-safe).
//  - One wave32 per B row: lane L handles columns [4L, 4L+3] as a 128-bit load.
//    B is streamed exactly once -> non-temporal load (preserve L2 for `out`,
//    which at 51.2 MB is fully resident in MI455X's 192 MB L2, so all atomic
//    RMW traffic stays on-chip).
//  - Target row index scalarized with readfirstlane; 4 native non-returning
//    global_atomic_add_f32 per lane (512 contiguous bytes per wave).

typedef float vfloat4 __attribute__((ext_vector_type(4)));  // native clang vector

__global__ __launch_bounds__(256) void scatter_add_rows_kernel(
    const int* __restrict__ index,       // [M] target row ids
    const vfloat4* __restrict__ B4,      // [M * 32] (128 floats per row)
    float* __restrict__ out,             // [N * 128]
    int M)
{
    const int lane        = threadIdx.x & 31;
    const int waveInBlock = threadIdx.x >> 5;           // 8 waves per 256-thread block
    const int row         = blockIdx.x * 8 + waveInBlock;
    if (row >= M) return;

    // Wave-uniform target row (scalarize the address math).
    const int target = __builtin_amdgcn_readfirstlane(index[row]);

    // Streaming read of B: non-temporal (TH=NT) so it doesn't evict `out` from L2.
    const vfloat4 b = __builtin_nontemporal_load(&B4[(size_t)row * 32 + lane]);

    float* o = out + (size_t)target * 128 + (size_t)lane * 4;

    // Native fp32 global atomics; return value unused -> non-returning form
    // (STOREcnt path), RMW resolved at L2.
    unsafeAtomicAdd(o + 0, b.x);
    unsafeAtomicAdd(o + 1, b.y);
    unsafeAtomicAdd(o + 2, b.z);
    unsafeAtomicAdd(o + 3, b.w);
}

extern "C" void kernel_launch(void* const* d_in, const int* in_sizes, int n_in,
                              void* d_out, int out_size, void* d_ws, size_t ws_size,
                              hipStream_t stream) {
    const int*     index = (const int*)    d_in[0];  // [M]
    const float*   A     = (const float*)  d_in[1];  // [N*128]
    const vfloat4* B4    = (const vfloat4*)d_in[2];  // [M*128] viewed as 4-float vectors
    float*         out   = (float*)d_out;            // [N*128]

    const int M = in_sizes[0];

    // 1) out = A  (async D2D copy on the capture stream)
    (void)hipMemcpyAsync(d_out, (const void*)A, (size_t)out_size * sizeof(float),
                         hipMemcpyDeviceToDevice, stream);

    // 2) scatter-add B rows into out (one wave32 per row, 8 waves/block)
    const int wavesPerBlock = 8;
    const int blocks = (M + wavesPerBlock - 1) / wavesPerBlock;
    scatter_add_rows_kernel<<<blocks, 256, 0, stream>>>(index, B4, out, M);
}